// OcclusionThirdLayer_66597762892551
// MI455X (gfx1250) — compile-verified
//
#include <hip/hip_runtime.h>

typedef __attribute__((ext_vector_type(2))) float v2f;
typedef __attribute__((ext_vector_type(8))) float v8f;

#define IMG_SIZE_IN   192      // 2*IMAGE_H
#define IMG_SIZE_OUT  18432    // 2*IMAGE_H*IMAGE_W
#define IMG_EPS_IN    9216     // IMAGE_H*IMAGE_W

// One wave computes 16 consecutive output rows:
//   out[r] = dot(W[r,:], x) + dot(Weps[r,:], eps) + bias[r] + bias_eps[r]
// using V_WMMA_F32_16X16X4_F32 with the vector broadcast across all 16 B-columns.
__global__ __launch_bounds__(256) void occlusion_gemv_wmma(
    const float* __restrict__ x,
    const float* __restrict__ eps,
    const float* __restrict__ W,
    const float* __restrict__ bias,
    const float* __restrict__ Weps,
    const float* __restrict__ bias_eps,
    float* __restrict__ out)
{
    const int lane = threadIdx.x & 31;
    const int wave = (int)(blockIdx.x * (blockDim.x >> 5) + (threadIdx.x >> 5));
    const int row0 = wave * 16;                  // 16 output rows per wave
    const int m    = lane & 15;                  // A-matrix row owned by this lane
    const int koff = (lane >> 4) << 1;           // K sub-offset: 0 for lanes 0-15, 2 for 16-31

    v8f c0 = {};   // two accumulators to break the WMMA D->C dependency chain
    v8f c1 = {};

    // ---- W_eps @ eps : K = 9216, step 8 (2 x k=4 WMMA) ----
    {
        const float* __restrict__ arow = Weps + (size_t)(row0 + m) * IMG_EPS_IN + koff;
        const float* __restrict__ brow = eps + koff;
        #pragma unroll 2
        for (int k = 0; k < IMG_EPS_IN; k += 8) {
            v2f a0 = *(const v2f*)(arow + k);
            v2f b0 = *(const v2f*)(brow + k);
            v2f a1 = *(const v2f*)(arow + k + 4);
            v2f b1 = *(const v2f*)(brow + k + 4);
            c0 = __builtin_amdgcn_wmma_f32_16x16x4_f32(
                false, a0, false, b0, (short)0, c0, false, false);
            c1 = __builtin_amdgcn_wmma_f32_16x16x4_f32(
                false, a1, false, b1, (short)0, c1, false, false);
        }
    }

    // ---- W @ x : K = 192, step 8 ----
    {
        const float* __restrict__ arow = W + (size_t)(row0 + m) * IMG_SIZE_IN + koff;
        const float* __restrict__ brow = x + koff;
        #pragma unroll
        for (int k = 0; k < IMG_SIZE_IN; k += 8) {
            v2f a0 = *(const v2f*)(arow + k);
            v2f b0 = *(const v2f*)(brow + k);
            v2f a1 = *(const v2f*)(arow + k + 4);
            v2f b1 = *(const v2f*)(brow + k + 4);
            c0 = __builtin_amdgcn_wmma_f32_16x16x4_f32(
                false, a0, false, b0, (short)0, c0, false, false);
            c1 = __builtin_amdgcn_wmma_f32_16x16x4_f32(
                false, a1, false, b1, (short)0, c1, false, false);
        }
    }

    v8f c;
    #pragma unroll
    for (int j = 0; j < 8; ++j) c[j] = c0[j] + c1[j];

    // D layout (16x16 f32 C/D): lanes 0-15 hold rows 0-7 in c[0..7] (every column
    // identical since B was broadcast), lanes 16-31 hold rows 8-15.
    // Lane 0 writes rows 0-7, lane 16 writes rows 8-15, adding both biases.
    if (m == 0) {
        const int base = row0 + ((lane >> 4) << 3);
        #pragma unroll
        for (int j = 0; j < 8; ++j) {
            out[base + j] = c[j] + bias[base + j] + bias_eps[base + j];
        }
    }
}

extern "C" void kernel_launch(void* const* d_in, const int* in_sizes, int n_in,
                              void* d_out, int out_size, void* d_ws, size_t ws_size,
                              hipStream_t stream) {
    const float* x        = (const float*)d_in[0];  // [192]
    const float* epsilon  = (const float*)d_in[1];  // [9216]
    const float* weights  = (const float*)d_in[2];  // [18432, 192]
    const float* bias     = (const float*)d_in[3];  // [18432]
    const float* weps     = (const float*)d_in[4];  // [18432, 9216]
    const float* bias_eps = (const float*)d_in[5];  // [18432]
    float* out            = (float*)d_out;          // [18432]

    // 18432 rows / 16 rows-per-wave = 1152 waves; 8 waves (256 thr) per block.
    const int waves  = IMG_SIZE_OUT / 16;
    const int blocks = waves / 8;
    occlusion_gemv_wmma<<<blocks, 256, 0, stream>>>(
        x, epsilon, weights, bias, weps, bias_eps, out);
}